// DenseLayerWithMask_39298950758467
// MI455X (gfx1250) — compile-verified
//
#include <hip/hip_runtime.h>
#include <hip/hip_bf16.h>

typedef float v2f __attribute__((ext_vector_type(2)));
typedef float v8f __attribute__((ext_vector_type(8)));

#define BATCH   4096
#define DIM     4096      // IN_DIM == OUT_DIM
#define NSPLIT  64
#define DPS     64        // features per split (in and out)
#define BN_EPS  1e-3f
#define TSTRIDE 68        // padded LDS stride for transposed W block
                          // bank = (4n + k) % 64 -> conflict-free b64 loads

// ---------------------------------------------------------------------------
// Kernel 0: zero the stats accumulators in workspace
// ---------------------------------------------------------------------------
__global__ void dlm_zero_kernel(float* __restrict__ p, int n) {
    int i = blockIdx.x * 256 + threadIdx.x;
    if (i < n) p[i] = 0.0f;
}

// ---------------------------------------------------------------------------
// Kernel 1: block-diagonal GEMM + bias, write pre-BN y, accumulate column
//           sum / sum-of-squares for batch-norm statistics.
// grid = (NSPLIT, BATCH/128), block = 256 (8 waves); wave w handles rows
// [mtile*128 + w*16, +16) x all 64 columns of its split.
// W block is staged TRANSPOSED in LDS: lds_wT[n * TSTRIDE + k] = W[k][n],
// so each lane's B fragment {W[k][n], W[k+1][n]} is one contiguous
// ds_load_b64 straight into an even-aligned VGPR pair for WMMA.
// ---------------------------------------------------------------------------
__global__ __launch_bounds__(256)
void dlm_gemm_kernel(const float* __restrict__ x,
                     const float* __restrict__ w,
                     const float* __restrict__ bias,
                     float* __restrict__ y,
                     float* __restrict__ gsum,
                     float* __restrict__ gsumsq) {
    __shared__ __align__(16) float lds_wT[DPS * TSTRIDE];  // W^T, padded rows
    __shared__ float lds_s1[DPS];
    __shared__ float lds_s2[DPS];

    const int s     = blockIdx.x;       // split index
    const int mtile = blockIdx.y;       // 128-row tile
    const int tid   = threadIdx.x;
    const int wave  = tid >> 5;
    const int lane  = tid & 31;

    if (tid < DPS) { lds_s1[tid] = 0.0f; lds_s2[tid] = 0.0f; }

    // Cooperative load of the 64x64 diagonal W block, stored transposed.
    {
        const float* wblk = w + (size_t)(s * DPS) * DIM + s * DPS;
        for (int i = tid; i < (DPS * DPS / 4); i += 256) {
            const int r  = i >> 4;            // W row  (k) 0..63
            const int c4 = (i & 15) << 2;     // W cols (n) 0..60 step 4
            const float4 v = *(const float4*)(wblk + (size_t)r * DIM + c4);
            lds_wT[(c4 + 0) * TSTRIDE + r] = v.x;
            lds_wT[(c4 + 1) * TSTRIDE + r] = v.y;
            lds_wT[(c4 + 2) * TSTRIDE + r] = v.z;
            lds_wT[(c4 + 3) * TSTRIDE + r] = v.w;
        }
    }
    __syncthreads();

    const int m0    = mtile * 128 + wave * 16;
    const int mrow  = lane & 15;            // M (for A) / N (for B,C,D)
    const int khalf = (lane >> 4) << 1;     // 0 or 2: K sub-column pair

    const float* xrow = x + (size_t)(m0 + mrow) * DIM + s * DPS;

    v8f acc[4] = {};                        // four 16x16 N-tiles

    #pragma unroll 4
    for (int k = 0; k < DPS; k += 4) {
        // A fragment: 16x4 f32.  lane<16: K=k..k+1, lane>=16: K=k+2..k+3
        const v2f a = *(const v2f*)(xrow + k + khalf);
        #pragma unroll
        for (int nt = 0; nt < 4; ++nt) {
            const int n = nt * 16 + mrow;
            // B fragment: contiguous {W[k+khalf][n], W[k+khalf+1][n]}
            const v2f b = *(const v2f*)(&lds_wT[n * TSTRIDE + k + khalf]);
            acc[nt] = __builtin_amdgcn_wmma_f32_16x16x4_f32(
                false, a, false, b, (short)0, acc[nt], false, false);
        }
    }

    // bias + store pre-BN output + per-column partial stats
    const int rbase = m0 + ((lane >> 4) << 3);   // rows rbase..rbase+7
    #pragma unroll
    for (int nt = 0; nt < 4; ++nt) {
        const int n   = nt * 16 + mrow;
        const int col = s * DPS + n;
        const float bv = bias[col];
        float s1 = 0.0f, s2 = 0.0f;
        #pragma unroll
        for (int v = 0; v < 8; ++v) {
            const float val = acc[nt][v] + bv;
            y[(size_t)(rbase + v) * DIM + col] = val;
            s1 += val;
            s2 += val * val;
        }
        atomicAdd(&lds_s1[n], s1);
        atomicAdd(&lds_s2[n], s2);
    }
    __syncthreads();

    if (tid < DPS) {
        atomicAdd(&gsum[s * DPS + tid],   lds_s1[tid]);
        atomicAdd(&gsumsq[s * DPS + tid], lds_s2[tid]);
    }
}

// ---------------------------------------------------------------------------
// Kernel 2: fold sums into per-column BN scale/shift
// ---------------------------------------------------------------------------
__global__ void dlm_stats_kernel(const float* __restrict__ gsum,
                                 const float* __restrict__ gsumsq,
                                 const float* __restrict__ gamma,
                                 const float* __restrict__ beta,
                                 float* __restrict__ scale,
                                 float* __restrict__ shift) {
    const int c = blockIdx.x * 256 + threadIdx.x;
    if (c >= DIM) return;
    const float inv_b = 1.0f / (float)BATCH;
    const float mean  = gsum[c] * inv_b;
    const float var   = gsumsq[c] * inv_b - mean * mean;
    const float sc    = gamma[c] * rsqrtf(var + BN_EPS);
    scale[c] = sc;
    shift[c] = beta[c] - mean * sc;
}

// ---------------------------------------------------------------------------
// Kernel 3: in-place BN apply + ReLU, float4 vectorized
// ---------------------------------------------------------------------------
__global__ __launch_bounds__(256)
void dlm_bnrelu_kernel(float* __restrict__ y,
                       const float* __restrict__ scale,
                       const float* __restrict__ shift) {
    const size_t i    = (size_t)blockIdx.x * 256 + threadIdx.x;
    const size_t base = i * 4;
    const int col     = (int)(base & (DIM - 1));
    float4 v        = *(float4*)(y + base);
    const float4 sc = *(const float4*)(scale + col);
    const float4 sh = *(const float4*)(shift + col);
    v.x = fmaxf(fmaf(v.x, sc.x, sh.x), 0.0f);
    v.y = fmaxf(fmaf(v.y, sc.y, sh.y), 0.0f);
    v.z = fmaxf(fmaf(v.z, sc.z, sh.z), 0.0f);
    v.w = fmaxf(fmaf(v.w, sc.w, sh.w), 0.0f);
    *(float4*)(y + base) = v;
}

// ---------------------------------------------------------------------------
extern "C" void kernel_launch(void* const* d_in, const int* in_sizes, int n_in,
                              void* d_out, int out_size, void* d_ws, size_t ws_size,
                              hipStream_t stream) {
    (void)in_sizes; (void)n_in; (void)out_size; (void)ws_size;

    const float* x     = (const float*)d_in[0];
    const float* w     = (const float*)d_in[1];
    const float* bias  = (const float*)d_in[2];
    const float* gamma = (const float*)d_in[3];
    const float* beta  = (const float*)d_in[4];

    float* y  = (float*)d_out;
    float* ws = (float*)d_ws;
    float* gsum   = ws;             // [4096]
    float* gsumsq = ws + DIM;       // [4096]
    float* scale  = ws + 2 * DIM;   // [4096]
    float* shift  = ws + 3 * DIM;   // [4096]

    // 0) zero the atomic accumulators (ws is poisoned / carries old state)
    dlm_zero_kernel<<<(2 * DIM + 255) / 256, 256, 0, stream>>>(gsum, 2 * DIM);

    // 1) block-diagonal GEMM + bias + stats
    dim3 g1(NSPLIT, BATCH / 128);
    dlm_gemm_kernel<<<g1, 256, 0, stream>>>(x, w, bias, y, gsum, gsumsq);

    // 2) finalize BN scale/shift
    dlm_stats_kernel<<<DIM / 256, 256, 0, stream>>>(gsum, gsumsq, gamma, beta,
                                                    scale, shift);

    // 3) apply BN + ReLU in place
    const int n4 = (BATCH * DIM) / 4;
    dlm_bnrelu_kernel<<<n4 / 256, 256, 0, stream>>>(y, scale, shift);
}